// SAGELayer_66503273611812
// MI455X (gfx1250) — compile-verified
//
#include <hip/hip_runtime.h>

#define IN_F  128
#define OUT_F 128
// Transposed-W LDS stride: even (keeps 8B alignment of (k,k+1) pairs for even k)
// and not a multiple of 64 floats (spreads LDS banks across columns).
#define WT_STRIDE 130

typedef float v2f __attribute__((ext_vector_type(2)));
typedef float v8f __attribute__((ext_vector_type(8)));

// ---------------------------------------------------------------------------
// Kernel 1: h = x @ W + b  using V_WMMA_F32_16X16X4_F32 (fp32 matrix core).
// Block = 256 threads = 8 waves. Each wave owns one 16-row m-tile and sweeps
// all 8 n-tiles (A fragment reused 8x per k-step).
//  - W is staged TRANSPOSED in LDS (Wl[col*130 + k]) so each lane's B
//    fragment (k, k+1) is one aligned ds_load_b64 -> no repacking movs.
//  - The wave's whole A tile (32 x v2f per lane) is prefetched into VGPRs
//    before the k-loop, so no global-load waits sit on the WMMA critical path.
// ---------------------------------------------------------------------------
__global__ __launch_bounds__(256) void sage_gemm_wmma(
    const float* __restrict__ x, const float* __restrict__ W,
    const float* __restrict__ bias, float* __restrict__ h, int n_nodes) {
  __shared__ float Wl[OUT_F * WT_STRIDE];  // ~65 KB, well under 320 KB/WGP

  // Cooperative transpose of W into LDS: W[k][c] -> Wl[c*WT_STRIDE + k].
  // Each thread reads one coalesced float4 row-chunk and scatters 4 floats.
  {
    const int total4 = IN_F * OUT_F / 4;  // 4096 float4 chunks
    for (int i = threadIdx.x; i < total4; i += blockDim.x) {
      const int k = i >> 5;          // row of W (128 rows, 32 chunks each)
      const int c = (i & 31) * 4;    // starting column
      const float4 v = ((const float4*)W)[i];
      Wl[(c + 0) * WT_STRIDE + k] = v.x;
      Wl[(c + 1) * WT_STRIDE + k] = v.y;
      Wl[(c + 2) * WT_STRIDE + k] = v.z;
      Wl[(c + 3) * WT_STRIDE + k] = v.w;
    }
  }
  __syncthreads();

  const int waveId = threadIdx.x >> 5;
  const int lane   = threadIdx.x & 31;
  const int mtile  = blockIdx.x * 8 + waveId;
  if (mtile * 16 >= n_nodes) return;  // n_nodes % 16 == 0 for this problem
  const int m0   = mtile * 16;
  const int mrow = lane & 15;          // A: row within tile
  const int koff = (lane >> 4) << 1;   // A/B: lanes 16-31 carry K+2,K+3
  const int ncol = lane & 15;          // B/C/D: column within tile

  // Prefetch the full A tile for this wave: afrag[t] = x[row][4t+koff .. +1].
  const float* xrow = x + (size_t)(m0 + mrow) * IN_F + koff;
  v2f afrag[IN_F / 4];
#pragma unroll
  for (int t = 0; t < IN_F / 4; ++t) {
    afrag[t] = *(const v2f*)(xrow + 4 * t);  // 8B-aligned global_load_b64
  }

  v8f acc[8] = {};  // 8 n-tiles of 16x16 f32 accumulators (8 VGPRs each)

  const float* wt = Wl + ncol * WT_STRIDE + koff;  // lane's B column base
  for (int k0 = 0; k0 < IN_F; k0 += 4) {
#pragma unroll
    for (int nt = 0; nt < 8; ++nt) {
      // B fragment (k0+koff, k0+koff+1) for column nt*16+ncol: one ds_load_b64.
      v2f bf = *(const v2f*)(wt + nt * 16 * WT_STRIDE + k0);
      acc[nt] = __builtin_amdgcn_wmma_f32_16x16x4_f32(
          /*neg_a=*/false, afrag[k0 >> 2], /*neg_b=*/false, bf,
          /*c_mod=*/(short)0, acc[nt], /*reuse_a=*/false, /*reuse_b=*/false);
    }
  }

  // D layout: VGPR v holds rows v (lanes 0-15) and v+8 (lanes 16-31).
  const int rbase = m0 + ((lane >> 4) << 3);
#pragma unroll
  for (int nt = 0; nt < 8; ++nt) {
    const int colg = nt * 16 + ncol;
    const float bb = bias[colg];
#pragma unroll
    for (int v = 0; v < 8; ++v) {
      h[(size_t)(rbase + v) * OUT_F + colg] = acc[nt][v] + bb;
    }
  }
}

// ---------------------------------------------------------------------------
// Kernel 2: edge scatter. One wave (32 lanes) per edge; each lane moves a
// float4 of the 128-wide message. h (51 MB) and out (51 MB) both fit in the
// 192 MB L2, so gathers hit L2 and the non-returning f32 atomics resolve at
// the L2 atomic units (no return traffic).
// ---------------------------------------------------------------------------
__global__ __launch_bounds__(256) void sage_scatter(
    const float* __restrict__ h, const int* __restrict__ erow,
    const int* __restrict__ ecol, float* __restrict__ out,
    float* __restrict__ deg, int n_edges) {
  const int edge = (int)((blockIdx.x * (size_t)blockDim.x + threadIdx.x) >> 5);
  const int lane = threadIdx.x & 31;
  if (edge >= n_edges) return;
  const int r = erow[edge];
  const int c = ecol[edge];
  const float4 v = *(const float4*)(h + (size_t)c * OUT_F + lane * 4);
  float* o = out + (size_t)r * OUT_F + lane * 4;
  unsafeAtomicAdd(o + 0, v.x);
  unsafeAtomicAdd(o + 1, v.y);
  unsafeAtomicAdd(o + 2, v.z);
  unsafeAtomicAdd(o + 3, v.w);
  if (lane == 0) unsafeAtomicAdd(deg + r, 1.0f);
}

// ---------------------------------------------------------------------------
// Kernel 3: out = (agg/max(deg,1)) then L2-normalize. One wave per node,
// wave32 butterfly reduction for the squared norm.
// ---------------------------------------------------------------------------
__global__ __launch_bounds__(256) void sage_finalize(
    float* __restrict__ out, const float* __restrict__ deg, int n_nodes) {
  const int node = (int)((blockIdx.x * (size_t)blockDim.x + threadIdx.x) >> 5);
  const int lane = threadIdx.x & 31;
  if (node >= n_nodes) return;

  const float invd = 1.0f / fmaxf(deg[node], 1.0f);
  float4* p = (float4*)(out + (size_t)node * OUT_F + lane * 4);
  float4 v = *p;
  v.x *= invd; v.y *= invd; v.z *= invd; v.w *= invd;

  float ss = v.x * v.x + v.y * v.y + v.z * v.z + v.w * v.w;
#pragma unroll
  for (int off = 16; off > 0; off >>= 1) ss += __shfl_xor(ss, off, 32);

  const float s = 1.0f / fmaxf(sqrtf(ss), 1e-12f);
  v.x *= s; v.y *= s; v.z *= s; v.w *= s;
  *p = v;
}

// ---------------------------------------------------------------------------
// Launch. Inputs (setup_inputs order): x[100000*128] f32, edge_row[1.6M] i32,
// edge_col[1.6M] i32, W[128*128] f32, b[128] f32. Output: [100000*128] f32.
// Workspace: deg (n floats) then h (n*128 floats), ~51.6 MB total.
// ---------------------------------------------------------------------------
extern "C" void kernel_launch(void* const* d_in, const int* in_sizes, int n_in,
                              void* d_out, int out_size, void* d_ws, size_t ws_size,
                              hipStream_t stream) {
  const float* x    = (const float*)d_in[0];
  const int*   erow = (const int*)d_in[1];
  const int*   ecol = (const int*)d_in[2];
  const float* W    = (const float*)d_in[3];
  const float* bias = (const float*)d_in[4];
  float* out = (float*)d_out;

  const int n_nodes = in_sizes[0] / IN_F;
  const int n_edges = in_sizes[1];

  float* deg = (float*)d_ws;
  size_t h_off = (((size_t)n_nodes * sizeof(float)) + 255) & ~(size_t)255;
  float* h = (float*)((char*)d_ws + h_off);

  // Zero accumulators (graph-capture-safe async memsets).
  hipMemsetAsync(out, 0, (size_t)out_size * sizeof(float), stream);
  hipMemsetAsync(deg, 0, (size_t)n_nodes * sizeof(float), stream);

  const int mtiles = (n_nodes + 15) / 16;
  sage_gemm_wmma<<<(mtiles + 7) / 8, 256, 0, stream>>>(x, W, bias, h, n_nodes);
  sage_scatter<<<(n_edges + 7) / 8, 256, 0, stream>>>(h, erow, ecol, out, deg, n_edges);
  sage_finalize<<<(n_nodes + 7) / 8, 256, 0, stream>>>(out, deg, n_nodes);
}